// LogSpaceMinGRU_60318520705469
// MI455X (gfx1250) — compile-verified
//
#include <hip/hip_runtime.h>
#include <hip/hip_bf16.h>
#include <math.h>

// Problem constants (from reference setup_inputs): B=4, S=4096, D=1024.
#define B_DIM   4
#define S_LEN   4096
#define D_DIM   1024
#define M_TOTAL (B_DIM * S_LEN)      // 16384 rows of x
#define NCHUNK  32                   // chunks along S for the two-pass scan
#define CLEN    (S_LEN / NCHUNK)     // 128
#define NCHAN   (B_DIM * D_DIM)      // 4096 independent scan channels

// GEMM tiling
#define BM      128                  // block M tile (8 waves x 16)
#define BD      32                   // block d tile (hidden cols; + matching gate cols)
#define BK      32                   // K tile
#define LDSK    36                   // padded LDS row stride (16B-aligned rows, conflict-free frags)

#if __has_builtin(__builtin_amdgcn_global_load_async_to_lds_b128)
#define HAVE_ASYNC_LDS 1
#else
#define HAVE_ASYNC_LDS 0
#endif

typedef __attribute__((ext_vector_type(2))) float v2f;
typedef __attribute__((ext_vector_type(8))) float v8f;
typedef __attribute__((__vector_size__(4 * sizeof(int)))) int v4i;
typedef __attribute__((address_space(1))) v4i* as1_v4i_p;   // global v4i*
typedef __attribute__((address_space(3))) v4i* as3_v4i_p;   // LDS v4i*

// stable softplus(x) = log(1 + e^x)
__device__ __forceinline__ float softplusf(float x) {
    return fmaxf(x, 0.0f) + log1pf(__expf(-fabsf(x)));
}

// stable logaddexp with -inf handling
__device__ __forceinline__ float logaddexpf(float a, float b) {
    float m = fmaxf(a, b);
    if (m == -INFINITY) return -INFINITY;
    return m + log1pf(__expf(fminf(a, b) - m));
}

__device__ __forceinline__ void wait_async_and_barrier() {
#if HAVE_ASYNC_LDS
#if __has_builtin(__builtin_amdgcn_s_wait_asynccnt)
    __builtin_amdgcn_s_wait_asynccnt(0);
#else
    asm volatile("s_wait_asynccnt 0" ::: "memory");
#endif
#endif
    __syncthreads();
}

// -------------------------------------------------------------------------
// Kernel 1: fused GEMM (fp32 WMMA, async LDS staging, double-buffered)
//   hg[m, n] = sum_k x[m,k] * W[n,k]       (n < D: hidden, n >= D: gate)
//   logc = -softplus(gate)
//   logv = -softplus(-gate) + log_g(hidden)
// Block = 256 threads = 8 waves; block tile 128M x 32d (and the paired gate
// columns 1024+d), K staged through LDS in 32-wide tiles via
// GLOBAL_LOAD_ASYNC_TO_LDS_B128 (ASYNCcnt), overlapped with WMMA compute.
// -------------------------------------------------------------------------
__global__ __launch_bounds__(256) void gemm_logspace_kernel(
    const float* __restrict__ x,   // [M_TOTAL, D_DIM]
    const float* __restrict__ W,   // [2*D_DIM, D_DIM]
    float* __restrict__ logc,      // [M_TOTAL, D_DIM]
    float* __restrict__ logv)      // [M_TOTAL, D_DIM]
{
    __shared__ float Xs[2][BM * LDSK];        // 2 x 18 KB
    __shared__ float Ws[2][2 * BD * LDSK];    // 2 x  9 KB

    const int tid   = threadIdx.x;
    const int lane  = tid & 31;
    const int wave  = tid >> 5;
    const int dtile = blockIdx.x * BD;               // first hidden column
    const int mblk  = blockIdx.y * BM;               // first M row of block
    const int mbase = mblk + wave * 16;              // first M row of this wave

    // cooperative-load coordinates: 8 threads cover one 32-float row segment
    const int lrow = tid >> 3;                       // 0..31
    const int lcol = (tid & 7) * 4;                  // 0,4,...,28

    // fragment coordinates (A 16x4 f32 layout: lanes 0-15 K={0,1}, 16-31 K={2,3})
    const int row   = lane & 15;
    const int khalf = (lane >> 4) * 2;

    v8f ch0 = {}, ch1 = {}, cg0 = {}, cg1 = {};

    // stage one K-tile into LDS buffer `buf` starting at global column k0
    auto stage_tile = [&](int buf, int k0) {
#if HAVE_ASYNC_LDS
        #pragma unroll
        for (int i = 0; i < 4; ++i)
            __builtin_amdgcn_global_load_async_to_lds_b128(
                (as1_v4i_p)(x + (size_t)(mblk + lrow + 32 * i) * D_DIM + k0 + lcol),
                (as3_v4i_p)&Xs[buf][(lrow + 32 * i) * LDSK + lcol], 0, 0);
        __builtin_amdgcn_global_load_async_to_lds_b128(
            (as1_v4i_p)(W + (size_t)(dtile + lrow) * D_DIM + k0 + lcol),
            (as3_v4i_p)&Ws[buf][lrow * LDSK + lcol], 0, 0);
        __builtin_amdgcn_global_load_async_to_lds_b128(
            (as1_v4i_p)(W + (size_t)(D_DIM + dtile + lrow) * D_DIM + k0 + lcol),
            (as3_v4i_p)&Ws[buf][(32 + lrow) * LDSK + lcol], 0, 0);
#else
        float4 xr[4], wr[2];
        #pragma unroll
        for (int i = 0; i < 4; ++i)
            xr[i] = *(const float4*)(x + (size_t)(mblk + lrow + 32 * i) * D_DIM + k0 + lcol);
        wr[0] = *(const float4*)(W + (size_t)(dtile + lrow) * D_DIM + k0 + lcol);
        wr[1] = *(const float4*)(W + (size_t)(D_DIM + dtile + lrow) * D_DIM + k0 + lcol);
        #pragma unroll
        for (int i = 0; i < 4; ++i)
            *(float4*)&Xs[buf][(lrow + 32 * i) * LDSK + lcol] = xr[i];
        *(float4*)&Ws[buf][lrow * LDSK + lcol]        = wr[0];
        *(float4*)&Ws[buf][(32 + lrow) * LDSK + lcol] = wr[1];
#endif
    };

    // ---- prologue: stage K-tile 0 into buffer 0 ----
    stage_tile(0, 0);
    wait_async_and_barrier();

    for (int kt = 0; kt < D_DIM / BK; ++kt) {
        const int knext = (kt + 1) * BK;
        const int cur   = kt & 1;
        const int nxt   = cur ^ 1;

        // warm L2 two tiles ahead (global_prefetch_b8)
        const int kpf = (kt + 2) * BK;
        if (kpf < D_DIM) {
            __builtin_prefetch(x + (size_t)(mblk + lrow) * D_DIM + kpf + lcol, 0, 0);
            __builtin_prefetch(W + (size_t)(dtile + lrow) * D_DIM + kpf + lcol, 0, 0);
        }

        // kick off the next tile's cache->LDS DMA while computing current tile
        if (knext < D_DIM)
            stage_tile(nxt, knext);

        // ---- compute current K-tile from LDS ----
        const float* Xc = &Xs[cur][0];
        const float* Wc = &Ws[cur][0];
        #pragma unroll
        for (int kk = 0; kk < BK; kk += 4) {
            v2f a   = *(const v2f*)&Xc[(wave * 16 + row) * LDSK + kk + khalf];
            v2f bh0 = *(const v2f*)&Wc[(row)      * LDSK + kk + khalf];
            v2f bh1 = *(const v2f*)&Wc[(16 + row) * LDSK + kk + khalf];
            v2f bg0 = *(const v2f*)&Wc[(32 + row) * LDSK + kk + khalf];
            v2f bg1 = *(const v2f*)&Wc[(48 + row) * LDSK + kk + khalf];
            // (neg_a, A, neg_b, B, c_mod, C, reuse_a, reuse_b)
            ch0 = __builtin_amdgcn_wmma_f32_16x16x4_f32(false, a, false, bh0, (short)0, ch0, false, false);
            ch1 = __builtin_amdgcn_wmma_f32_16x16x4_f32(false, a, false, bh1, (short)0, ch1, false, false);
            cg0 = __builtin_amdgcn_wmma_f32_16x16x4_f32(false, a, false, bg0, (short)0, cg0, false, false);
            cg1 = __builtin_amdgcn_wmma_f32_16x16x4_f32(false, a, false, bg1, (short)0, cg1, false, false);
        }

        // make next buffer visible (per-wave async wait + block barrier)
        if (knext < D_DIM)
            wait_async_and_barrier();
    }

    // C/D layout: VGPR r, lanes 0-15 -> M=r, lanes 16-31 -> M=r+8; N = lane&15
    const int mrow_off = (lane >> 4) * 8;
    const int dcol     = dtile + (lane & 15);

    #pragma unroll
    for (int r = 0; r < 8; ++r) {
        size_t base = (size_t)(mbase + mrow_off + r) * D_DIM;
        {
            float h = ch0[r], g = cg0[r];
            float lc  = -softplusf(g);
            float lz  = -softplusf(-g);
            float lgh = (h >= 0.0f) ? __logf(h + 0.5f) : -softplusf(-h);
            logc[base + dcol]      = lc;
            logv[base + dcol]      = lz + lgh;
        }
        {
            float h = ch1[r], g = cg1[r];
            float lc  = -softplusf(g);
            float lz  = -softplusf(-g);
            float lgh = (h >= 0.0f) ? __logf(h + 0.5f) : -softplusf(-h);
            logc[base + dcol + 16] = lc;
            logv[base + dcol + 16] = lz + lgh;
        }
    }
}

// -------------------------------------------------------------------------
// Kernel 2 (pass A): per (channel, chunk) local scan with h_in = -inf.
// Produces per-chunk decay A_c = sum(logc) and local state L_c.
// -------------------------------------------------------------------------
__global__ __launch_bounds__(256) void scan_passA_kernel(
    const float* __restrict__ logc, const float* __restrict__ logv,
    float* __restrict__ chunkA, float* __restrict__ chunkL)
{
    int t  = blockIdx.x * blockDim.x + threadIdx.x;   // [0, NCHUNK*NCHAN)
    int ch = t % NCHAN;
    int c  = t / NCHAN;
    int b  = ch / D_DIM;
    int d  = ch % D_DIM;

    size_t base = ((size_t)b * S_LEN + (size_t)c * CLEN) * D_DIM + d;
    float A = 0.0f;
    float H = -INFINITY;
    for (int s = 0; s < CLEN; ++s) {
        float lc = logc[base + (size_t)s * D_DIM];
        float lv = logv[base + (size_t)s * D_DIM];
        A += lc;
        H = logaddexpf(lc + H, lv);
    }
    chunkA[(size_t)c * NCHAN + ch] = A;
    chunkL[(size_t)c * NCHAN + ch] = H;
}

// -------------------------------------------------------------------------
// Kernel 3: cross-chunk serial scan (tiny: 4096 channels x 32 chunks).
// carry[c][ch] = log-state entering chunk c.
// -------------------------------------------------------------------------
__global__ __launch_bounds__(256) void scan_mid_kernel(
    const float* __restrict__ chunkA, const float* __restrict__ chunkL,
    float* __restrict__ carry)
{
    int ch = blockIdx.x * blockDim.x + threadIdx.x;   // [0, NCHAN)
    float H = -INFINITY;
    for (int c = 0; c < NCHUNK; ++c) {
        carry[(size_t)c * NCHAN + ch] = H;
        float A = chunkA[(size_t)c * NCHAN + ch];
        float L = chunkL[(size_t)c * NCHAN + ch];
        H = logaddexpf(A + H, L);
    }
}

// -------------------------------------------------------------------------
// Kernel 4 (pass B): re-scan each chunk with the true incoming state,
// emit out = exp(log_h).
// -------------------------------------------------------------------------
__global__ __launch_bounds__(256) void scan_passB_kernel(
    const float* __restrict__ logc, const float* __restrict__ logv,
    const float* __restrict__ carry, float* __restrict__ out)
{
    int t  = blockIdx.x * blockDim.x + threadIdx.x;
    int ch = t % NCHAN;
    int c  = t / NCHAN;
    int b  = ch / D_DIM;
    int d  = ch % D_DIM;

    size_t base = ((size_t)b * S_LEN + (size_t)c * CLEN) * D_DIM + d;
    float H = carry[(size_t)c * NCHAN + ch];
    for (int s = 0; s < CLEN; ++s) {
        float lc = logc[base + (size_t)s * D_DIM];
        float lv = logv[base + (size_t)s * D_DIM];
        H = logaddexpf(lc + H, lv);
        out[base + (size_t)s * D_DIM] = __expf(H);
    }
}

// -------------------------------------------------------------------------
// Launch
// -------------------------------------------------------------------------
extern "C" void kernel_launch(void* const* d_in, const int* in_sizes, int n_in,
                              void* d_out, int out_size, void* d_ws, size_t ws_size,
                              hipStream_t stream) {
    (void)in_sizes; (void)n_in; (void)out_size; (void)ws_size;

    const float* x = (const float*)d_in[0];   // [B, S, D]
    const float* W = (const float*)d_in[1];   // [2D, D]
    float* out = (float*)d_out;               // [B, S, D]

    // Workspace layout (floats): logc | logv | chunkA | chunkL | carry
    float* logc   = (float*)d_ws;
    float* logv   = logc   + (size_t)M_TOTAL * D_DIM;
    float* chunkA = logv   + (size_t)M_TOTAL * D_DIM;
    float* chunkL = chunkA + (size_t)NCHUNK * NCHAN;
    float* carry  = chunkL + (size_t)NCHUNK * NCHAN;

    // 1) Fused WMMA GEMM + log-space epilogue
    dim3 gblock(256);
    dim3 ggrid(D_DIM / BD, M_TOTAL / BM);
    gemm_logspace_kernel<<<ggrid, gblock, 0, stream>>>(x, W, logc, logv);

    // 2) Chunked two-pass log-space scan along S
    int nA = NCHUNK * NCHAN;                  // 131072 threads
    scan_passA_kernel<<<nA / 256, 256, 0, stream>>>(logc, logv, chunkA, chunkL);
    scan_mid_kernel<<<NCHAN / 256, 256, 0, stream>>>(chunkA, chunkL, carry);
    scan_passB_kernel<<<nA / 256, 256, 0, stream>>>(logc, logv, carry, out);
}